// InterNodeMoELayerOut_2199023256088
// MI455X (gfx1250) — compile-verified
//
#include <hip/hip_runtime.h>
#include <hip/hip_bf16.h>
#include <stdint.h>

// MoE un-scatter/combine for MI455X (gfx1250, wave32).
// Pure streaming kernel: ~270 MB HBM traffic, ~33 MFLOP -> HBM-bound (~11.5 us
// floor @ 23.3 TB/s). No contraction dim -> WMMA inapplicable; we instead use
// CDNA5's async global->LDS copy path for the indirect expert-row gather and
// non-temporal b128 loads/stores so the 288 MB working set does not churn the
// 192 MB L2.

typedef float v4f __attribute__((ext_vector_type(4)));

#ifndef USE_ASYNC_LDS
#define USE_ASYNC_LDS 1
#endif

__device__ __forceinline__ void wait_async0() {
#if __has_builtin(__builtin_amdgcn_s_wait_asynccnt)
    __builtin_amdgcn_s_wait_asynccnt(0);
#else
    asm volatile("s_wait_asynccnt 0x0" ::: "memory");
#endif
}

template <int THREADS>
__global__ __launch_bounds__(THREADS) void moe_combine_kernel(
    const float* __restrict__ eo,            // [EC, D] expert_outputs
    const float* __restrict__ x,             // [T, D]  residual tokens
    const float* __restrict__ rpm,           // [T]     route_prob_max
    const int* __restrict__ sidx,            // [EC]    scatter_idx (T = padding)
    const unsigned char* __restrict__ drop,  // [T]     jnp bool -> 1 byte/elem
    float* __restrict__ out,                 // [T, D]
    int EC, int T, int D)
{
    const int b   = (int)blockIdx.x;
    const int tid = (int)threadIdx.x;
    const int D4  = D >> 2;  // row length in float4

    if (b < EC) {
        // ---- scatter slot: out[s,:] = eo[b,:] * rpm[s] ----
        const int s = sidx[b];                      // uniform -> scalar load
        if ((unsigned)s >= (unsigned)T) return;     // capacity padding slot
        const float f = rpm[s];
        const v4f* __restrict__ src = (const v4f*)(eo + (size_t)b * (size_t)D);
        v4f* __restrict__ dst       = (v4f*)(out + (size_t)s * (size_t)D);

#if USE_ASYNC_LDS
        __shared__ __align__(16) float stage[THREADS * 4];
        for (int c0 = 0; c0 < D4; c0 += THREADS) {
            const int c = c0 + tid;
            if (c < D4) {
                // CDNA5 async copy: GLOBAL_LOAD_ASYNC_TO_LDS_B128
                // VDST = LDS byte offset (vgpr), VADDR = 64-bit address pair.
                uint32_t lds_b = (uint32_t)(uintptr_t)(&stage[4 * tid]);
                uint64_t g     = (uint64_t)(uintptr_t)(src + c);
                asm volatile("global_load_async_to_lds_b128 %0, %1, off"
                             :
                             : "v"(lds_b), "v"(g)
                             : "memory");
            }
            wait_async0();  // s_wait_asynccnt 0: our lane's 16B are in LDS
            if (c < D4) {
                // each lane reads back only the slot it loaded -> no barrier
                v4f v = *(const v4f*)(&stage[4 * tid]);
                v = v * f;
                __builtin_nontemporal_store(v, dst + c);  // th:NT, write-once
            }
            if (c0 + THREADS < D4) __syncthreads();  // WAR on stage (not hit at D=2048)
        }
#else
        for (int c = tid; c < D4; c += THREADS) {
            v4f v = __builtin_nontemporal_load(src + c);
            v = v * f;
            __builtin_nontemporal_store(v, dst + c);
        }
#endif
    } else {
        // ---- dropped token: out[t,:] = x[t,:] (factor 1.0) ----
        const int t = b - EC;
        if (!drop[t]) return;  // kept tokens are written by their scatter slot
        const v4f* __restrict__ src = (const v4f*)(x + (size_t)t * (size_t)D);
        v4f* __restrict__ dst       = (v4f*)(out + (size_t)t * (size_t)D);
        for (int c = tid; c < D4; c += THREADS) {
            v4f v = __builtin_nontemporal_load(src + c);
            __builtin_nontemporal_store(v, dst + c);
        }
    }
}

extern "C" void kernel_launch(void* const* d_in, const int* in_sizes, int n_in,
                              void* d_out, int out_size, void* d_ws, size_t ws_size,
                              hipStream_t stream) {
    const float* eo           = (const float*)d_in[0];          // expert_outputs
    const float* x            = (const float*)d_in[1];          // x
    const float* rpm          = (const float*)d_in[2];          // route_prob_max
    const int* sidx           = (const int*)d_in[3];            // scatter_idx
    const unsigned char* drop = (const unsigned char*)d_in[4];  // dropped_mask (bool)
    float* out                = (float*)d_out;

    const int EC = in_sizes[3];           // E*C = 20480
    const int T  = in_sizes[2];           // 16384
    const int D  = in_sizes[0] / EC;      // 2048

    constexpr int THREADS = 512;          // 16 wave32 / block; 1 float4 per lane per row
    const int grid = EC + T;              // one block per slot + one per token

    hipLaunchKernelGGL((moe_combine_kernel<THREADS>), dim3(grid), dim3(THREADS), 0,
                       stream, eo, x, rpm, sidx, drop, out, EC, T, D);
}